// CAPELoss_89292370084481
// MI455X (gfx1250) — compile-verified
//
#include <hip/hip_runtime.h>
#include <hip/hip_bf16.h>
#include <stdint.h>

// Problem constants (from reference)
#define Bc      2
#define Dd      128
#define Hh      256
#define Ww      256
#define NVOX    (Dd * Hh * Ww)     // 8388608
#define NPAIRS  32
#define NITERS  10
#define SMOOTHF 1e-5f

// Pool tile: 16(W) x 16(H) x 8(D) outputs per block, 256 threads = 8 waves,
// one 16x16 output z-slice per wave (matches WMMA 16x16 D-tile exactly).
#define TX 16
#define TY 16
#define TZ 8
#define HALO_N ((TZ + 2) * (TY + 2) * (TX + 2))   // 10*18*18 = 3240
#define WSUM_N ((TZ + 2) * (TY + 2) * TX)         // 10*18*16 = 2880

typedef __attribute__((ext_vector_type(2))) float v2f;
typedef __attribute__((ext_vector_type(8))) float v8f;

// ---------------------------------------------------------------------------
// CDNA5 async global->LDS staging (ASYNCcnt path), guarded with fallback.
// ---------------------------------------------------------------------------
typedef __attribute__((address_space(1))) int as1_int;
typedef __attribute__((address_space(3))) int as3_int;

__device__ __forceinline__ void async_ld_f32(float* lds_dst, const float* gsrc) {
#if __has_builtin(__builtin_amdgcn_global_load_async_to_lds_b32)
  __builtin_amdgcn_global_load_async_to_lds_b32(
      (as1_int*)gsrc, (as3_int*)lds_dst, 0, 0);
#else
  *lds_dst = *gsrc;
#endif
}

__device__ __forceinline__ void async_wait_all() {
#if __has_builtin(__builtin_amdgcn_s_wait_asynccnt)
  __builtin_amdgcn_s_wait_asynccnt(0);
#elif __has_builtin(__builtin_amdgcn_global_load_async_to_lds_b32)
  asm volatile("s_wait_asynccnt 0" ::: "memory");
#endif
}

// ---------------------------------------------------------------------------
// fg_prob = softmax(logits, axis=C)[1] = 1/(1+exp(l0-l1)), vectorized x4
// ---------------------------------------------------------------------------
__global__ __launch_bounds__(256) void softmax_kernel(
    const float* __restrict__ l0, const float* __restrict__ l1,
    float* __restrict__ prob) {
  size_t i = ((size_t)blockIdx.x * 256 + threadIdx.x) * 4;
  float4 a = *(const float4*)(l0 + i);
  float4 b = *(const float4*)(l1 + i);
  float4 p;
  p.x = __fdividef(1.0f, 1.0f + __expf(a.x - b.x));
  p.y = __fdividef(1.0f, 1.0f + __expf(a.y - b.y));
  p.z = __fdividef(1.0f, 1.0f + __expf(a.z - b.z));
  p.w = __fdividef(1.0f, 1.0f + __expf(a.w - b.w));
  *(float4*)(prob + i) = p;
}

// Zero the heat buffer (vectorized) + the per-iteration max slots
__global__ __launch_bounds__(256) void zero_heat_kernel(
    float* __restrict__ t0, unsigned int* __restrict__ maxraw) {
  size_t i = ((size_t)blockIdx.x * 256 + threadIdx.x) * 4;
  *(float4*)(t0 + i) = make_float4(0.f, 0.f, 0.f, 0.f);
  if (blockIdx.x == 0 && threadIdx.x < 16) maxraw[threadIdx.x] = 0u;
}

// Scatter 1.0 at endpoints_a (coords are (d,h,w))
__global__ void scatter_kernel(float* __restrict__ t0, const int* __restrict__ ea) {
  int j = threadIdx.x;  // NPAIRS threads
  int d = ea[3 * j + 0], h = ea[3 * j + 1], w = ea[3 * j + 2];
  t0[((size_t)d * Hh + h) * Ww + w] = 1.0f;
}

__global__ void zero_out_kernel(float* __restrict__ out) { out[0] = 0.0f; }

// ---------------------------------------------------------------------------
// One unnormalized iteration: tout = avgpool3d_3x3x3(tin) * prob, plus the
// iteration max (u32 atomic-max of non-negative float bits).
//
// Pipeline per block:
//   A) async global->LDS halo stage (18x18x10)
//   B) 3-tap sum along W in LDS -> wsum[10][18][16]
//   C) per wave w: 16x16 output slice z=z0+w,
//        D = sum_{dz=0..2} T16 x wsum[w+dz]   via 12 chained
//        v_wmma_f32_16x16x4_f32 ops (T16 = banded 0/1 matrix, built in regs),
//        + 2-row VALU fixup for the 2 halo rows cut off by K=16,
//        then * (1/27) * prob, store, track max.
// ---------------------------------------------------------------------------
__global__ __launch_bounds__(256) void pool_kernel(
    const float* __restrict__ tin, float* __restrict__ tout,
    const float* __restrict__ prob, unsigned int* __restrict__ maxdst) {
  __shared__ float tile[TZ + 2][TY + 2][TX + 2];
  __shared__ float wsum[TZ + 2][TY + 2][TX];
  __shared__ unsigned int smax;

  const int tid = threadIdx.x;
  const int x0 = blockIdx.x * TX, y0 = blockIdx.y * TY, z0 = blockIdx.z * TZ;
  if (tid == 0) smax = 0u;

  // Stage A: async halo staging (zeros outside the volume)
  for (int t = tid; t < HALO_N; t += 256) {
    int lz = t / ((TY + 2) * (TX + 2));
    int r  = t % ((TY + 2) * (TX + 2));
    int ly = r / (TX + 2);
    int lx = r % (TX + 2);
    int gz = z0 + lz - 1, gy = y0 + ly - 1, gx = x0 + lx - 1;
    float* ldst = &tile[lz][ly][lx];
    if ((unsigned)gz < (unsigned)Dd && (unsigned)gy < (unsigned)Hh &&
        (unsigned)gx < (unsigned)Ww) {
      async_ld_f32(ldst, tin + ((size_t)gz * Hh + gy) * Ww + gx);
    } else {
      *ldst = 0.0f;
    }
  }

  const int lane = tid & 31;
  const int wv   = tid >> 5;      // wave id = output z-slice within tile
  const int grp  = lane >> 4;     // lane group (0: lanes 0-15, 1: lanes 16-31)
  const int M    = lane & 15;     // WMMA A row, also B / D column

  // Prefetch prob rows for this wave's slice while async copies are in flight
#pragma unroll
  for (int r = 0; r < 8; r++) {
    __builtin_prefetch(
        prob + ((size_t)(z0 + wv) * Hh + (y0 + r + 8 * grp)) * Ww + (x0 + M), 0, 0);
  }

  async_wait_all();
  __syncthreads();

  // Stage B: 3-tap sum along W for all 18 halo rows of all 10 slices
  for (int t = tid; t < WSUM_N; t += 256) {
    int lz = t / ((TY + 2) * TX);
    int r  = t % ((TY + 2) * TX);
    int ly = r / TX;
    int lx = r % TX;
    wsum[lz][ly][lx] = tile[lz][ly][lx] + tile[lz][ly][lx + 1] + tile[lz][ly][lx + 2];
  }
  __syncthreads();

  // Stage C
  v8f acc = {};

#if __has_builtin(__builtin_amdgcn_wmma_f32_16x16x4_f32)
  // A operand: T16[i][j] = 1 if i <= j <= i+2 (j in 0..15), built per-lane.
  // A layout (16x4 f32): lane l, vgpr v -> A[M = l%16][K = v + 2*(l/16)]
  v2f av[4];
#pragma unroll
  for (int c = 0; c < 4; c++) {
    int j0 = 4 * c + 2 * grp;
    av[c].x = (j0 >= M && j0 <= M + 2) ? 1.0f : 0.0f;
    av[c].y = (j0 + 1 >= M && j0 + 1 <= M + 2) ? 1.0f : 0.0f;
  }
  // B layout (4x16 f32): lane l, vgpr v -> B[K = v + 2*(l/16)][N = l%16]
#pragma unroll
  for (int dz = 0; dz < 3; dz++) {
    const int lz = wv + dz;
#pragma unroll
    for (int c = 0; c < 4; c++) {
      v2f bv;
      int row0 = 4 * c + 2 * grp;
      bv.x = wsum[lz][row0 + 0][M];
      bv.y = wsum[lz][row0 + 1][M];
      acc = __builtin_amdgcn_wmma_f32_16x16x4_f32(
          false, av[c], false, bv, (short)0, acc, false, false);
    }
  }
  // Edge fixup: K=16 dropped halo rows j=16,17 (y'=15,16). They contribute to
  // output rows 14 (j=16) and 15 (j=16,17) only -> r=6,7 in lane group 1.
  if (grp == 1) {
    float f16s = 0.0f, f17s = 0.0f;
#pragma unroll
    for (int dz = 0; dz < 3; dz++) {
      f16s += wsum[wv + dz][16][M];
      f17s += wsum[wv + dz][17][M];
    }
    acc[6] += f16s;
    acc[7] += f16s + f17s;
  }
#else
  // VALU fallback: direct 9-point (H,D) sum
#pragma unroll
  for (int r = 0; r < 8; r++) {
    int my = r + 8 * grp;
    float s = 0.0f;
#pragma unroll
    for (int dz = 0; dz < 3; dz++)
#pragma unroll
      for (int dy = 0; dy < 3; dy++)
        s += wsum[wv + dz][my + dy][M];
    acc[r] = s;
  }
#endif

  // Scale, fuse prob, store, track max.
  // D layout (16x16 f32): vgpr r, lane l -> D[Mrow = r + 8*(l/16)][N = l%16]
  float lmax = 0.0f;
  const int gz = z0 + wv;
#pragma unroll
  for (int r = 0; r < 8; r++) {
    int gy   = y0 + r + 8 * grp;
    size_t g = ((size_t)gz * Hh + gy) * Ww + (x0 + M);
    float v  = acc[r] * (1.0f / 27.0f) * prob[g];
    tout[g]  = v;
    lmax     = fmaxf(lmax, v);
  }
  atomicMax(&smax, __float_as_uint(lmax));
  __syncthreads();
  if (tid == 0) atomicMax(maxdst, smax);
}

// ---------------------------------------------------------------------------
// Final gather: fold the 10 deferred normalizations into a scalar recurrence
// s_k = s_{k-1} / (s_{k-1}*M_k + eps), then loss_b = 1 - mean(u10[eb]*s_10).
// Deterministic: single thread does the sequential reduction.
// ---------------------------------------------------------------------------
__global__ void gather_kernel(const float* __restrict__ t10,
                              const unsigned int* __restrict__ maxraw,
                              const int* __restrict__ eb,
                              float* __restrict__ out) {
  __shared__ float vals[NPAIRS];
  int j = threadIdx.x;  // NPAIRS threads
  int d = eb[3 * j + 0], h = eb[3 * j + 1], w = eb[3 * j + 2];
  vals[j] = t10[((size_t)d * Hh + h) * Ww + w];
  __syncthreads();
  if (j == 0) {
    float s = 1.0f;
    for (int k = 0; k < NITERS; k++) {
      float M = __uint_as_float(maxraw[k]);
      if (M > 0.0f) s = s / (s * M + SMOOTHF);  // mirrors where(hmax>0, ...)
    }
    float acc = 0.0f;
    for (int i = 0; i < NPAIRS; i++) acc += vals[i];
    float loss = 1.0f - (acc * (1.0f / NPAIRS)) * s;
    out[0] += loss * (1.0f / Bc);
  }
}

// ---------------------------------------------------------------------------
// Launch: batches processed SEQUENTIALLY so the ~100MB per-batch working set
// (prob + ping-pong heat) stays resident in the MI455X's 192MB L2 across the
// 10-iteration dependent loop.
// ---------------------------------------------------------------------------
extern "C" void kernel_launch(void* const* d_in, const int* in_sizes, int n_in,
                              void* d_out, int out_size, void* d_ws, size_t ws_size,
                              hipStream_t stream) {
  (void)in_sizes; (void)n_in; (void)out_size; (void)ws_size;
  const float* logits = (const float*)d_in[0];
  // d_in[1] = labels (unused: endpoints are precomputed inputs)
  const int* ea = (const int*)d_in[2];
  const int* eb = (const int*)d_in[3];
  float* out = (float*)d_out;

  float* prob = (float*)d_ws;                 // NVOX floats
  float* t0   = prob + NVOX;                  // NVOX floats (ping)
  float* t1   = t0 + NVOX;                    // NVOX floats (pong)
  unsigned int* maxraw = (unsigned int*)(t1 + NVOX);  // 16 u32 slots

  zero_out_kernel<<<1, 1, 0, stream>>>(out);

  dim3 pgrid(Ww / TX, Hh / TY, Dd / TZ);      // (16,16,16)
  for (int b = 0; b < Bc; b++) {
    const float* l0 = logits + (size_t)b * 2 * NVOX;
    const float* l1 = l0 + NVOX;
    softmax_kernel<<<NVOX / 1024, 256, 0, stream>>>(l0, l1, prob);
    zero_heat_kernel<<<NVOX / 1024, 256, 0, stream>>>(t0, maxraw);
    scatter_kernel<<<1, NPAIRS, 0, stream>>>(t0, ea + b * NPAIRS * 3);

    float* tin = t0; float* tout = t1;
    for (int k = 0; k < NITERS; k++) {
      pool_kernel<<<pgrid, 256, 0, stream>>>(tin, tout, prob, &maxraw[k]);
      float* tmp = tin; tin = tout; tout = tmp;
    }
    gather_kernel<<<1, NPAIRS, 0, stream>>>(tin, maxraw, eb + b * NPAIRS * 3, out);
  }
}